// StableResonantSSM_61220463837187
// MI455X (gfx1250) — compile-verified
//
#include <hip/hip_runtime.h>

typedef float v2f __attribute__((ext_vector_type(2)));
typedef float v8f __attribute__((ext_vector_type(8)));

#define BATCH 8
#define SEQ   4096
#define DIM   768
#define NST   64
#define MTOT  (BATCH*SEQ)          // 32768

// workspace layout in floats
#define WS_BDC  0                      // 128*768   : [re(64 rows); im(64 rows)] of B_d
#define WS_AD   (WS_BDC + 128*DIM)     // 128       : a_re[64], a_im[64]
#define WS_BU   (WS_AD + 128)          // MTOT*128  : Bu (re cols 0..63, im cols 64..127)
#define WS_H    (WS_BU + MTOT*128)     // MTOT*64   : Re(h)

// ---------------------------------------------------------------------------
// Phase 0: bilinear discretization.  a_d = (1+dt/2 λ)/(1-dt/2 λ),
// B_d = sqrt(dt) * B / (1-dt/2 λ).   grid = 64 states, block = 768 dims.
// ---------------------------------------------------------------------------
__global__ void ssm_disc(const float* __restrict__ Ar, const float* __restrict__ Ai,
                         const float* __restrict__ Bm, float* __restrict__ ws) {
  const float dt = 0.01f, sdt = 0.1f;      // sqrt(0.01) = 0.1
  int n = blockIdx.x, d = threadIdx.x;
  float dr  = 1.0f - 0.5f * dt * Ar[n];    // denom = dr + i*di
  float di  = -0.5f * dt * Ai[n];
  float inv = 1.0f / (dr * dr + di * di);  // 1/denom = (dr - i*di)*inv
  float b   = Bm[n * DIM + d] * sdt;
  ws[WS_BDC + n * DIM + d]        =  b * dr * inv;
  ws[WS_BDC + (64 + n) * DIM + d] = -b * di * inv;
  if (d == 0) {
    float nr = 1.0f + 0.5f * dt * Ar[n];
    float ni = 0.5f * dt * Ai[n];
    ws[WS_AD + n]      = (nr * dr + ni * di) * inv;
    ws[WS_AD + 64 + n] = (ni * dr - nr * di) * inv;
  }
}

// ---------------------------------------------------------------------------
// GEMM1: Bu(32768x128) = u(32768x768) @ Bdc^T(768x128) via v_wmma_f32_16x16x4.
// Register-blocked: each wave owns a 64x16 panel (4 M-subtiles sharing one B
// fragment per K-step -> 5 b64 loads per 4 WMMAs).  8 waves/block cover all
// 128 output columns; block covers 64 rows.  grid = 512.
// ---------------------------------------------------------------------------
__global__ __launch_bounds__(256) void ssm_gemm1(const float* __restrict__ u,
                                                 const float* __restrict__ Bdc,
                                                 float* __restrict__ bu) {
  int m0   = blockIdx.x * 64;
  int wave = threadIdx.x >> 5;
  int lane = threadIdx.x & 31;
  int half = lane >> 4;
  int r    = lane & 15;
  int n0   = wave * 16;

  const float* a0 = u + (size_t)(m0 +      r) * DIM + 2 * half;
  const float* a1 = u + (size_t)(m0 + 16 + r) * DIM + 2 * half;
  const float* a2 = u + (size_t)(m0 + 32 + r) * DIM + 2 * half;
  const float* a3 = u + (size_t)(m0 + 48 + r) * DIM + 2 * half;
  const float* br = Bdc + (size_t)(n0 + r) * DIM + 2 * half;

  v8f c0 = {}, c1 = {}, c2 = {}, c3 = {};
#pragma unroll 2
  for (int k = 0; k < DIM; k += 4) {
    v2f b  = *(const v2f*)(br + k);
    v2f x0 = *(const v2f*)(a0 + k);
    v2f x1 = *(const v2f*)(a1 + k);
    v2f x2 = *(const v2f*)(a2 + k);
    v2f x3 = *(const v2f*)(a3 + k);
    c0 = __builtin_amdgcn_wmma_f32_16x16x4_f32(false, x0, false, b, (short)0, c0, false, false);
    c1 = __builtin_amdgcn_wmma_f32_16x16x4_f32(false, x1, false, b, (short)0, c1, false, false);
    c2 = __builtin_amdgcn_wmma_f32_16x16x4_f32(false, x2, false, b, (short)0, c2, false, false);
    c3 = __builtin_amdgcn_wmma_f32_16x16x4_f32(false, x3, false, b, (short)0, c3, false, false);
  }
  // D layout: row = mbase + v + 8*half, col = n0 + r
  float* o0 = bu + (size_t)(m0 +      8 * half) * 128 + n0 + r;
  float* o1 = bu + (size_t)(m0 + 16 + 8 * half) * 128 + n0 + r;
  float* o2 = bu + (size_t)(m0 + 32 + 8 * half) * 128 + n0 + r;
  float* o3 = bu + (size_t)(m0 + 48 + 8 * half) * 128 + n0 + r;
#pragma unroll
  for (int v = 0; v < 8; ++v) {
    o0[(size_t)v * 128] = c0[v];
    o1[(size_t)v * 128] = c1[v];
    o2[(size_t)v * 128] = c2[v];
    o3[(size_t)v * 128] = c3[v];
  }
}

// ---------------------------------------------------------------------------
// Scan: h_t = a*h_{t-1} + Bu_t (complex, constant a).  One block per (b,n);
// 128 threads x 32-step serial chunks; chunk carries combined by a
// Hillis-Steele scan in LDS with coefficient a^32 (repeated squaring).
// Writes only Re(h).
// ---------------------------------------------------------------------------
__global__ __launch_bounds__(128) void ssm_scan(const float* __restrict__ wa,
                                                const float* __restrict__ bu,
                                                float* __restrict__ h) {
  __shared__ float sr[128], si[128];
  int seq = blockIdx.x;                 // 0..511
  int b = seq >> 6, n = seq & 63;
  int t = threadIdx.x;                  // chunk id
  float ar = wa[n], ai = wa[64 + n];
  const float* base = bu + (size_t)b * SEQ * 128 + n;
  float*       hb   = h  + (size_t)b * SEQ * 64 + n;
  int s0 = t * 32;

  // pass 1: local scan with zero init -> chunk carry (lr, li)
  float lr = 0.f, li = 0.f;
  for (int k = 0; k < 32; ++k) {
    const float* p = base + (size_t)(s0 + k) * 128;
    float brv = p[0], biv = p[64];
    float nr = ar * lr - ai * li + brv;
    float ni = ar * li + ai * lr + biv;
    lr = nr; li = ni;
  }

  // q = a^32 via repeated squaring
  float qr = 1.f, qi = 0.f, pr = ar, pi = ai;
  for (int e = 32; e; e >>= 1) {
    if (e & 1) { float tr = qr * pr - qi * pi, ti = qr * pi + qi * pr; qr = tr; qi = ti; }
    float tr = pr * pr - pi * pi, ti = 2.f * pr * pi; pr = tr; pi = ti;
  }

  // inclusive Hillis-Steele scan of H[t] = q^off * H[t-off] + H[t]
  sr[t] = lr; si[t] = li;
  __syncthreads();
  float cr = qr, ci = qi, vr = lr, vi = li;
  for (int off = 1; off < 128; off <<= 1) {
    float xr = 0.f, xi = 0.f;
    if (t >= off) { xr = sr[t - off]; xi = si[t - off]; }
    __syncthreads();
    float tr = cr * xr - ci * xi + vr;
    float ti = cr * xi + ci * xr + vi;
    vr = tr; vi = ti;
    sr[t] = vr; si[t] = vi;
    __syncthreads();
    float ncr = cr * cr - ci * ci, nci = 2.f * cr * ci;  // coeff -> coeff^2
    cr = ncr; ci = nci;
  }

  // exclusive prefix = carry entering this chunk
  float hr = 0.f, hi = 0.f;
  if (t > 0) { hr = sr[t - 1]; hi = si[t - 1]; }

  // pass 2: rescan chunk with true carry, emit Re(h)
  for (int k = 0; k < 32; ++k) {
    const float* p = base + (size_t)(s0 + k) * 128;
    float brv = p[0], biv = p[64];
    float nr = ar * hr - ai * hi + brv;
    float ni = ar * hi + ai * hr + biv;
    hr = nr; hi = ni;
    hb[(size_t)(s0 + k) * 64] = hr;
  }
}

// ---------------------------------------------------------------------------
// GEMM2: y(32768x768) = h_re(32768x64) @ C^T(64x768).  Register-blocked the
// same way: 64x16 panel per wave, K=64 fully unrolled (16 WMMAs per subtile).
// grid = (512, 6); 8 waves/block -> 48 D-tiles of 16.
// ---------------------------------------------------------------------------
__global__ __launch_bounds__(256) void ssm_gemm2(const float* __restrict__ h,
                                                 const float* __restrict__ Cm,
                                                 float* __restrict__ y) {
  int m0   = blockIdx.x * 64;
  int wave = threadIdx.x >> 5;
  int lane = threadIdx.x & 31;
  int half = lane >> 4;
  int r    = lane & 15;
  int d0   = (blockIdx.y * 8 + wave) * 16;

  const float* a0 = h + (size_t)(m0 +      r) * 64 + 2 * half;
  const float* a1 = h + (size_t)(m0 + 16 + r) * 64 + 2 * half;
  const float* a2 = h + (size_t)(m0 + 32 + r) * 64 + 2 * half;
  const float* a3 = h + (size_t)(m0 + 48 + r) * 64 + 2 * half;
  const float* br = Cm + (size_t)(d0 + r) * 64 + 2 * half;  // B[k][j] = C[d0+j][k]

  v8f c0 = {}, c1 = {}, c2 = {}, c3 = {};
#pragma unroll
  for (int k = 0; k < 64; k += 4) {
    v2f b  = *(const v2f*)(br + k);
    v2f x0 = *(const v2f*)(a0 + k);
    v2f x1 = *(const v2f*)(a1 + k);
    v2f x2 = *(const v2f*)(a2 + k);
    v2f x3 = *(const v2f*)(a3 + k);
    c0 = __builtin_amdgcn_wmma_f32_16x16x4_f32(false, x0, false, b, (short)0, c0, false, false);
    c1 = __builtin_amdgcn_wmma_f32_16x16x4_f32(false, x1, false, b, (short)0, c1, false, false);
    c2 = __builtin_amdgcn_wmma_f32_16x16x4_f32(false, x2, false, b, (short)0, c2, false, false);
    c3 = __builtin_amdgcn_wmma_f32_16x16x4_f32(false, x3, false, b, (short)0, c3, false, false);
  }
  float* o0 = y + (size_t)(m0 +      8 * half) * DIM + d0 + r;
  float* o1 = y + (size_t)(m0 + 16 + 8 * half) * DIM + d0 + r;
  float* o2 = y + (size_t)(m0 + 32 + 8 * half) * DIM + d0 + r;
  float* o3 = y + (size_t)(m0 + 48 + 8 * half) * DIM + d0 + r;
#pragma unroll
  for (int v = 0; v < 8; ++v) {
    o0[(size_t)v * DIM] = c0[v];
    o1[(size_t)v * DIM] = c1[v];
    o2[(size_t)v * DIM] = c2[v];
    o3[(size_t)v * DIM] = c3[v];
  }
}

// ---------------------------------------------------------------------------
extern "C" void kernel_launch(void* const* d_in, const int* in_sizes, int n_in,
                              void* d_out, int out_size, void* d_ws, size_t ws_size,
                              hipStream_t stream) {
  (void)in_sizes; (void)n_in; (void)out_size; (void)ws_size;
  const float* u  = (const float*)d_in[0];
  const float* Ar = (const float*)d_in[1];
  const float* Ai = (const float*)d_in[2];
  const float* Bm = (const float*)d_in[3];
  const float* Cm = (const float*)d_in[4];
  float* ws = (float*)d_ws;
  float* bu = ws + WS_BU;
  float* h  = ws + WS_H;
  float* y  = (float*)d_out;

  ssm_disc <<<NST, DIM, 0, stream>>>(Ar, Ai, Bm, ws + WS_BDC);
  ssm_gemm1<<<MTOT / 64, 256, 0, stream>>>(u, ws + WS_BDC, bu);
  ssm_scan <<<BATCH * NST, 128, 0, stream>>>(ws + WS_AD, bu, h);
  ssm_gemm2<<<dim3(MTOT / 64, 6), 256, 0, stream>>>(h, Cm, y);
}